// GNN_24635932409979
// MI455X (gfx1250) — compile-verified
//
#include <hip/hip_runtime.h>
#include <hip/hip_bf16.h>

typedef __attribute__((ext_vector_type(2))) float v2f;
typedef __attribute__((ext_vector_type(8))) float v8f;

#define NHID 8
#define MAXN 2000

// ---------------- GCN stage ----------------

__global__ void k_start(const int* __restrict__ batch, int* __restrict__ start, int N, int B) {
    int n = blockIdx.x * blockDim.x + threadIdx.x;
    if (n >= N) return;
    int b = batch[n];
    if (n == 0 || batch[n - 1] != b) start[b] = n;
    if (n == N - 1) start[B] = N;
}

__global__ void k_deg_init(float* __restrict__ deg, int N) {
    int n = blockIdx.x * blockDim.x + threadIdx.x;
    if (n < N) deg[n] = 1.0f;   // self loop
}

__global__ void k_deg_acc(const int* __restrict__ dst, float* __restrict__ deg, int E) {
    int e = blockIdx.x * blockDim.x + threadIdx.x;
    if (e < E) atomicAdd(&deg[dst[e]], 1.0f);
}

__global__ void k_rsqrt(float* __restrict__ deg, int N) {
    int n = blockIdx.x * blockDim.x + threadIdx.x;
    if (n < N) deg[n] = rsqrtf(deg[n]);
}

// hpre[n,:] = hin[n,:] @ W   (Cin x NHID)
__global__ void k_xform(const float* __restrict__ hin, const float* __restrict__ W,
                        float* __restrict__ hpre, int N, int Cin) {
    int n = blockIdx.x * blockDim.x + threadIdx.x;
    if (n >= N) return;
    float acc[NHID];
#pragma unroll
    for (int c = 0; c < NHID; ++c) acc[c] = 0.0f;
    for (int i = 0; i < Cin; ++i) {
        float x = hin[(long long)n * Cin + i];
#pragma unroll
        for (int c = 0; c < NHID; ++c) acc[c] += x * W[i * NHID + c];
    }
#pragma unroll
    for (int c = 0; c < NHID; ++c) hpre[(long long)n * NHID + c] = acc[c];
}

// agg init with self-loop contribution
__global__ void k_selfinit(const float* __restrict__ hpre, const float* __restrict__ dinv,
                           float* __restrict__ agg, int N) {
    int n = blockIdx.x * blockDim.x + threadIdx.x;
    if (n >= N) return;
    float w = dinv[n] * dinv[n];
#pragma unroll
    for (int c = 0; c < NHID; ++c)
        agg[(long long)n * NHID + c] = hpre[(long long)n * NHID + c] * w;
}

__global__ void k_edge(const int* __restrict__ src, const int* __restrict__ dst,
                       const float* __restrict__ hpre, const float* __restrict__ dinv,
                       float* __restrict__ agg, int E) {
    int e = blockIdx.x * blockDim.x + threadIdx.x;
    if (e >= E) return;
    int s = src[e], d = dst[e];
    float w = dinv[s] * dinv[d];
#pragma unroll
    for (int c = 0; c < NHID; ++c)
        atomicAdd(&agg[(long long)d * NHID + c], hpre[(long long)s * NHID + c] * w);
}

__global__ void k_finish(const float* __restrict__ agg, const float* __restrict__ bias,
                         float* __restrict__ h, int N) {
    int n = blockIdx.x * blockDim.x + threadIdx.x;
    if (n >= N) return;
#pragma unroll
    for (int c = 0; c < NHID; ++c)
        h[(long long)n * NHID + c] = fmaxf(agg[(long long)n * NHID + c] + bias[c], 0.0f);
}

__global__ void k_zero(float* __restrict__ p, int n) {
    int i = blockIdx.x * blockDim.x + threadIdx.x;
    if (i < n) p[i] = 0.0f;
}

// dense[b][c][pos] = h[n][c]  (channel-major per graph, zero padded to MAXN)
__global__ void k_scatter(const float* __restrict__ h, const int* __restrict__ batch,
                          const int* __restrict__ start, float* __restrict__ dense, int N) {
    int n = blockIdx.x * blockDim.x + threadIdx.x;
    if (n >= N) return;
    int b = batch[n];
    int p = n - start[b];
    if (p >= MAXN) return;
#pragma unroll
    for (int c = 0; c < NHID; ++c)
        dense[((long long)b * NHID + c) * MAXN + p] = h[(long long)n * NHID + c];
}

// ---------------- Conv1d(k=5,'same') + bias + relu + pool2, via fp32 WMMA ----------------
// GEMM view per graph: Y[COUT, LIN] = W'[COUT, K] x im2col'[K, LIN],  K = CIN*5.
// K reordered tap-major: k' = t*CIN + i (valid since reduction order is free);
// weights permuted once into LDS, input slice staged in LDS with pad zeros baked in,
// so the inner loop is ds_load + v_wmma only (no bounds branches, no div-by-5).
//
// A frag (16x4 f32): lanes 0-15 -> M rows, VGPR0/1 = K0/K1; lanes 16-31 -> K2/K3.
// B frag (4x16 f32): lanes 0-15 -> N cols, VGPR0/1 = K0/K1; lanes 16-31 -> K2/K3.
// C/D (16x16 f32):   VGPR j: lanes 0-15 = (M=j, N=lane); lanes 16-31 = (M=j+8, N=lane-16).
template <int CIN, int COUT, int LIN>
__global__ void __launch_bounds__(128)
k_conv(const float* __restrict__ zin, const float* __restrict__ wgt,
       const float* __restrict__ bias, float* __restrict__ out) {
    constexpr int K = CIN * 5;
    constexpr int MT = COUT / 16;
    constexpr int LOUT = LIN / 2;
    constexpr int LG2CIN = (CIN == 8) ? 3 : (CIN == 16 ? 4 : 5);
    constexpr int SPAN = 68;             // 64 positions + 2 halo each side

    __shared__ float w_s[COUT * K];      // permuted: w_s[row*K + (t*CIN+i)]
    __shared__ float z_s[CIN * SPAN];    // input slice, zero-padded
    __shared__ float b_s[COUT];

    const int tid = threadIdx.x;
    const int b = blockIdx.y;
    const int l0b = blockIdx.x * 64;     // first output position of this block
    const float* z = zin + (long long)b * CIN * LIN;

    __builtin_prefetch(z + l0b, 0, 0);   // global_prefetch_b8 of upcoming slice

    // stage permuted weights
    for (int idx = tid; idx < COUT * K; idx += 128) {
        int row = idx / K;
        int kp  = idx - row * K;
        int t   = kp >> LG2CIN;
        int i   = kp & (CIN - 1);
        w_s[idx] = wgt[row * K + i * 5 + t];
    }
    for (int i = tid; i < COUT; i += 128) b_s[i] = bias[i];

    // stage input slice: z_s[i][c] = z[i][l0b - 2 + c], zero outside [0, LIN)
    for (int idx = tid; idx < CIN * SPAN; idx += 128) {
        int i = idx / SPAN;
        int c = idx - i * SPAN;
        int l = l0b - 2 + c;
        z_s[idx] = (l >= 0 && l < LIN) ? z[i * LIN + l] : 0.0f;
    }
    __syncthreads();

    const int wave = tid >> 5;
    const int lane = tid & 31;
    const int ntiles = (LIN + 15) / 16;
    const int nt = blockIdx.x * 4 + wave;
    if (nt >= ntiles) return;            // whole-wave uniform exit (EXEC all-1 for WMMA)

    const int half = lane >> 4;          // 0: K pair {0,1}, 1: K pair {2,3}
    const int r = lane & 15;             // M row (A) / N col (B)
    const int cbase = wave * 16 + r;     // column in z_s for tap t=0

    v8f acc[MT] = {};

    for (int k0 = 0; k0 < K; k0 += 4) {
        const int kb = k0 + 2 * half;    // even; kb%CIN even => kb,kb+1 share tap t
        const int t = kb >> LG2CIN;
        const int i = kb & (CIN - 1);
        const int c = cbase + t;
        v2f bf;
        bf[0] = z_s[i * SPAN + c];
        bf[1] = z_s[(i + 1) * SPAN + c];
#pragma unroll
        for (int m = 0; m < MT; ++m) {
            const int row = m * 16 + r;
            v2f af;
            af[0] = w_s[row * K + kb];
            af[1] = w_s[row * K + kb + 1];
            acc[m] = __builtin_amdgcn_wmma_f32_16x16x4_f32(
                false, af, false, bf, (short)0, acc[m], false, false);
        }
    }

    // bias + relu + pool2 (pair-average via lane shuffle), store [COUT, LOUT]
    const int l0 = nt * 16;
    float* outb = out + (long long)b * COUT * LOUT;
#pragma unroll
    for (int m = 0; m < MT; ++m) {
#pragma unroll
        for (int j = 0; j < 8; ++j) {
            int row = m * 16 + j + half * 8;
            float v = fmaxf(acc[m][j] + b_s[row], 0.0f);
            float v2 = __shfl_xor(v, 1, 32);
            if ((r & 1) == 0) {
                int l = l0 + r;
                if (l < LIN) outb[row * LOUT + (l >> 1)] = 0.5f * (v + v2);
            }
        }
    }
}

// ---------------- segment means -> flat [B, 640] ----------------
__global__ void k_segmean(const float* __restrict__ p3, const int* __restrict__ start,
                          float* __restrict__ flat) {
    int b = blockIdx.x;
    int c = threadIdx.x;                 // 0..63
    int cnt = start[b + 1] - start[b];
    int valid = cnt >> 3;                // num_nodes // 8
    int base = valid / 10;
    int rem = valid - base * 10;
    const float* row = p3 + ((long long)b * 64 + c) * 250;
    float* fb = flat + (long long)b * 640 + c * 10;
    int s = 0;
    for (int j = 0; j < 10; ++j) {
        int size = base + (j < rem ? 1 : 0);
        float sum = 0.0f;
        for (int p = s; p < s + size; ++p) sum += row[p];
        fb[j] = (size > 0) ? (sum / (float)size) : 0.0f;
        s += size;
    }
}

// ---------------- MLP head 640 -> 100 -> 2 ----------------
__global__ void k_fc(const float* __restrict__ flat, const float* __restrict__ fw1,
                     const float* __restrict__ fb1, const float* __restrict__ fw2,
                     const float* __restrict__ fb2, float* __restrict__ out) {
    int b = blockIdx.x;
    int t = threadIdx.x;                 // 128
    __shared__ float hid[100];
    const float* f = flat + (long long)b * 640;
    if (t < 100) {
        float s = fb1[t];
        for (int i = 0; i < 640; ++i) s += f[i] * fw1[i * 100 + t];
        hid[t] = fmaxf(s, 0.0f);
    }
    __syncthreads();
    if (t < 2) {
        float s = fb2[t];
        for (int j = 0; j < 100; ++j) s += hid[j] * fw2[j * 2 + t];
        out[b * 2 + t] = s;
    }
}

// ---------------- launch ----------------
extern "C" void kernel_launch(void* const* d_in, const int* in_sizes, int n_in,
                              void* d_out, int out_size, void* d_ws, size_t ws_size,
                              hipStream_t stream) {
    const float* x   = (const float*)d_in[0];
    const int*   ei  = (const int*)d_in[1];
    const int*   bat = (const int*)d_in[2];
    const float* W1  = (const float*)d_in[3];
    const float* b1  = (const float*)d_in[4];
    const float* W2  = (const float*)d_in[5];
    const float* b2  = (const float*)d_in[6];
    const float* cw1 = (const float*)d_in[7];
    const float* cb1 = (const float*)d_in[8];
    const float* cw2 = (const float*)d_in[9];
    const float* cb2 = (const float*)d_in[10];
    const float* cw3 = (const float*)d_in[11];
    const float* cb3 = (const float*)d_in[12];
    const float* fw1 = (const float*)d_in[13];
    const float* fb1 = (const float*)d_in[14];
    const float* fw2 = (const float*)d_in[15];
    const float* fb2 = (const float*)d_in[16];

    const int N = in_sizes[0] / 4;       // 339840
    const int E = in_sizes[1] / 2;       // edges
    const int B = out_size / 2;          // 256
    const long long N8 = (long long)N * NHID;
    const int DENSE = B * NHID * MAXN;   // 4,096,000 floats

    float* ws = (float*)d_ws;
    // layout (floats): [hpre 8N][agg 8N][h 8N][deg N][start pad 512][dense]
    float* f_hpre  = ws;
    float* f_agg   = ws + N8;
    float* f_h     = ws + 2 * N8;
    float* f_deg   = ws + 3 * N8;
    int*   i_start = (int*)(ws + 3 * N8 + N);
    float* f_dense = ws + 3 * N8 + N + 512;
    // reuse: p1/p3 over [hpre..agg) (4.096M <= 16N floats), p2 over dense, flat over h
    float* f_p1    = ws;
    float* f_p2    = f_dense;
    float* f_p3    = ws;
    float* f_flat  = f_h;

    const int T = 256;
    const int gN = (N + T - 1) / T;
    const int gE = (E + T - 1) / T;
    const int* src = ei;
    const int* dst = ei + E;

    // graph offsets + degree norm
    k_start<<<gN, T, 0, stream>>>(bat, i_start, N, B);
    k_deg_init<<<gN, T, 0, stream>>>(f_deg, N);
    k_deg_acc<<<gE, T, 0, stream>>>(dst, f_deg, E);
    k_rsqrt<<<gN, T, 0, stream>>>(f_deg, N);

    // GCN layer 1: x(4) -> h(8)
    k_xform<<<gN, T, 0, stream>>>(x, W1, f_hpre, N, 4);
    k_selfinit<<<gN, T, 0, stream>>>(f_hpre, f_deg, f_agg, N);
    k_edge<<<gE, T, 0, stream>>>(src, dst, f_hpre, f_deg, f_agg, E);
    k_finish<<<gN, T, 0, stream>>>(f_agg, b1, f_h, N);

    // GCN layer 2: h(8) -> h(8)
    k_xform<<<gN, T, 0, stream>>>(f_h, W2, f_hpre, N, 8);
    k_selfinit<<<gN, T, 0, stream>>>(f_hpre, f_deg, f_agg, N);
    k_edge<<<gE, T, 0, stream>>>(src, dst, f_hpre, f_deg, f_agg, E);
    k_finish<<<gN, T, 0, stream>>>(f_agg, b2, f_h, N);

    // dense batch [B,8,2000]
    k_zero<<<(DENSE + T - 1) / T, T, 0, stream>>>(f_dense, DENSE);
    k_scatter<<<gN, T, 0, stream>>>(f_h, bat, i_start, f_dense, N);

    // conv stack (WMMA f32): [B,8,2000]->[B,16,1000]->[B,32,500]->[B,64,250]
    {
        dim3 g((2000 + 63) / 64, B);
        k_conv<8, 16, 2000><<<g, 128, 0, stream>>>(f_dense, cw1, cb1, f_p1);
    }
    {
        dim3 g((1000 + 63) / 64, B);
        k_conv<16, 32, 1000><<<g, 128, 0, stream>>>(f_p1, cw2, cb2, f_p2);
    }
    {
        dim3 g((500 + 63) / 64, B);
        k_conv<32, 64, 500><<<g, 128, 0, stream>>>(f_p2, cw3, cb3, f_p3);
    }

    // segment means + MLP head
    k_segmean<<<B, 64, 0, stream>>>(f_p3, i_start, f_flat);
    k_fc<<<B, 128, 0, stream>>>(f_flat, fw1, fb1, fw2, fb2, (float*)d_out);
}